// NeRFRenderer_54228257080081
// MI455X (gfx1250) — compile-verified
//
#include <hip/hip_runtime.h>
#include <hip/hip_bf16.h>
#include <stdint.h>

#define BLOCK 64
#define NUM_STEPS 128
#define UP_STEPS 128

// ---- TDM descriptor helper types (amdgpu-toolchain 6-arg builtin form) ----
typedef unsigned int tdm_u32x4 __attribute__((ext_vector_type(4)));
typedef int          tdm_i32x8 __attribute__((ext_vector_type(8)));
typedef int          tdm_i32x4 __attribute__((ext_vector_type(4)));

// D# group 0: count=1 (user), lds_addr, 57-bit global addr, type=2 ("image")
__device__ __forceinline__ tdm_u32x4 tdm_group0(uint32_t lds_addr, uint64_t gaddr) {
    tdm_u32x4 g;
    g[0] = 1u;                                               // count=1, is_restore=0, gather off
    g[1] = lds_addr;                                         // LDS byte address
    g[2] = (uint32_t)gaddr;                                  // global_addr[31:0]
    g[3] = ((uint32_t)(gaddr >> 32) & 0x01FFFFFFu) | (2u << 30); // global_addr[56:32] | type=2
    return g;
}

// D# group 1: 1-D tile of n 4-byte elements, no multicast/pad/iterate/barrier.
__device__ __forceinline__ tdm_i32x8 tdm_group1_1d(uint32_t n) {
    tdm_i32x8 g;
    g[0] = (int)(2u << 16);                       // wg_mask=0 | data_size=2 (4B)
    g[1] = (int)((n & 0xFFFFu) << 16);            // abar_addr=0 | tensor_dim0[15:0]
    g[2] = (int)((n >> 16) | (1u << 16));         // tensor_dim0[31:16] | tensor_dim1=1 (lo)
    g[3] = (int)((n & 0xFFFFu) << 16);            // tensor_dim1 hi=0 | tile_dim0=n
    g[4] = 1;                                     // tile_dim1=1 | tile_dim2=0
    g[5] = (int)n;                                // tensor_dim0_stride lo32
    g[6] = (int)((n & 0xFFFFu) << 16);            // stride0 hi=0 | tensor_dim1_stride lo16
    g[7] = 0;                                     // tensor_dim1_stride hi
    return g;
}

__device__ __forceinline__ float sigmoidf_(float x) {
    return 1.0f / (1.0f + __expf(-x));
}

__global__ __launch_bounds__(BLOCK) void nerf_render_kernel(
    const float* __restrict__ rays_o, const float* __restrict__ rays_d,
    const float* __restrict__ cptr,   const float* __restrict__ log_s,
    const float* __restrict__ amp,    const float* __restrict__ Wcp,
    const float* __restrict__ bcp,    float* __restrict__ out)
{
    __shared__ float s_rays[6 * BLOCK];            // [0,3B): rays_o rows; [3B,6B): rays_d rows
    __shared__ float s_newz[UP_STEPS * BLOCK];     // new_z[k] for thread t at [k*BLOCK + t]

    const int tid  = threadIdx.x;
    const int ray0 = blockIdx.x * BLOCK;

    // ---- Stage this block's ray data into LDS via the Tensor Data Mover ----
    if (tid < 32) {  // wave 0 only (wave32); TDM ignores EXEC, one issue per wave
        const uint32_t n4 = 3u * BLOCK;  // 192 dwords per tile
        uint64_t ga_o = (uint64_t)(uintptr_t)(rays_o + (size_t)ray0 * 3);
        uint64_t ga_d = (uint64_t)(uintptr_t)(rays_d + (size_t)ray0 * 3);
        uint32_t ld_o = (uint32_t)(uintptr_t)(&s_rays[0]);
        uint32_t ld_d = (uint32_t)(uintptr_t)(&s_rays[3 * BLOCK]);
        tdm_i32x8 g1 = tdm_group1_1d(n4);
        tdm_i32x4 z4 = {0, 0, 0, 0};
        tdm_i32x8 z8 = {0, 0, 0, 0, 0, 0, 0, 0};
        __builtin_amdgcn_tensor_load_to_lds(tdm_group0(ld_o, ga_o), g1, z4, z4, z8, 0);
        __builtin_amdgcn_tensor_load_to_lds(tdm_group0(ld_d, ga_d), g1, z4, z4, z8, 0);
        __builtin_amdgcn_s_wait_tensorcnt(0);
    }
    __syncthreads();

    // ---- Small params (uniform; served from cache) ----
    const float cx = cptr[0], cy = cptr[1], cz = cptr[2];
    const float sA = __expf(amp[0]);
    const float ss = __expf(log_s[0]);
    const float inv2s2 = 0.5f / (ss * ss);
    float W[9];
#pragma unroll
    for (int i = 0; i < 9; ++i) W[i] = Wcp[i];
    const float b0 = bcp[0], b1 = bcp[1], b2 = bcp[2];

    // ---- Ray setup ----
    float ox = s_rays[3 * tid + 0], oy = s_rays[3 * tid + 1], oz = s_rays[3 * tid + 2];
    float dx = s_rays[3 * BLOCK + 3 * tid + 0];
    float dy = s_rays[3 * BLOCK + 3 * tid + 1];
    float dz = s_rays[3 * BLOCK + 3 * tid + 2];
    {
        float invn = 1.0f / sqrtf(dx * dx + dy * dy + dz * dz);
        dx *= invn; dy *= invn; dz *= invn;
    }

    // near/far from AABB [-1,1]^3
    float tmin = -3.0e38f, tmax = 3.0e38f;
    {
        float dv[3] = {dx, dy, dz}, ov[3] = {ox, oy, oz};
#pragma unroll
        for (int a = 0; a < 3; ++a) {
            float dd = dv[a];
            float sd = (fabsf(dd) > 1e-8f)
                       ? dd
                       : ((dd > 0.f ? 1e-8f : (dd < 0.f ? -1e-8f : 0.f)) + 1e-12f);
            float inv = 1.0f / sd;
            float t1 = (-1.0f - ov[a]) * inv;
            float t2 = ( 1.0f - ov[a]) * inv;
            tmin = fmaxf(tmin, fminf(t1, t2));
            tmax = fminf(tmax, fmaxf(t1, t2));
        }
    }
    const float nearz = fmaxf(tmin, 0.05f);
    const float farz  = fmaxf(tmax, nearz + 1e-4f);
    const float span  = farz - nearz;
    const float h     = span * (1.0f / 127.0f);     // coarse z spacing (first 127 deltas)
    const float sdist = span * (1.0f / 128.0f);     // sample_dist (last delta)

    // density at a point on the ray
    auto sigma_at = [&](float z) -> float {
        float x0 = fminf(fmaxf(ox + dx * z, -1.0f), 1.0f);
        float x1 = fminf(fmaxf(oy + dy * z, -1.0f), 1.0f);
        float x2 = fminf(fmaxf(oz + dz * z, -1.0f), 1.0f);
        float e0 = x0 - cx, e1 = x1 - cy, e2 = x2 - cz;
        float d2 = e0 * e0 + e1 * e1 + e2 * e2;
        return sA * __expf(-d2 * inv2s2);
    };

    // ---- Coarse pass A: sum of (w_i + 1e-5) over i = 1..126 (PDF normalizer) ----
    float T = 1.0f, wsum_pdf = 0.0f;
    for (int i = 0; i < NUM_STEPS; ++i) {
        float z = nearz + span * ((float)i * (1.0f / 127.0f));
        float sig = sigma_at(z);
        float delta = (i < 127) ? h : sdist;
        float alpha = 1.0f - __expf(-delta * sig);
        float w = alpha * T;
        T *= (1.0f - alpha + 1e-15f);
        if (i >= 1 && i <= 126) wsum_pdf += w + 1e-5f;
    }
    const float inv_wsum = 1.0f / wsum_pdf;  // >= 126e-5, safe

    // ---- Coarse pass B: streaming CDF + inverse-CDF sampling (u_k ascending) ----
    // u_k = 1/256 + k/128 ; searchsorted(cdf, u, 'right') realized as a 2-pointer merge.
    {
        // i = 0 only advances transmittance
        float sig0 = sigma_at(nearz);
        float a0 = 1.0f - __expf(-h * sig0);
        T = (1.0f - a0 + 1e-15f);

        float cprev = 0.0f;
        int k = 0;
        for (int i = 1; i <= 126; ++i) {
            float z = nearz + span * ((float)i * (1.0f / 127.0f));
            float sig = sigma_at(z);
            float alpha = 1.0f - __expf(-h * sig);
            float w = alpha * T;
            T *= (1.0f - alpha + 1e-15f);
            float pdf = (w + 1e-5f) * inv_wsum;
            float cnext = cprev + pdf;
            float binsb = nearz + h * ((float)(i - 1) + 0.5f);  // z_mid[i-1]
            float dinv  = (pdf < 1e-5f) ? 1.0f : (1.0f / pdf);  // denom<1e-5 -> 1 (ref)
            while (k < UP_STEPS) {
                float u = (1.0f / 256.0f) + (float)k * (1.0f / 128.0f);
                if (!(u < cnext)) break;          // ind == i  (cdf[i] > u, cdf[i-1] <= u)
                float t = (u - cprev) * dinv;
                s_newz[k * BLOCK + tid] = binsb + t * h;  // lerp(z_mid[i-1], z_mid[i])
                ++k;
            }
            cprev = cnext;
        }
        float zlast = nearz + h * 126.5f;  // ind==127 -> below=above=126 -> z_mid[126]
        while (k < UP_STEPS) { s_newz[k * BLOCK + tid] = zlast; ++k; }
    }
    __syncthreads();  // keep waves in step before LDS re-read phase

    // ---- Fine pass: streaming merge of uniform z and new_z, fused compositing ----
    int ia = 0, ib = 0;
    auto pop = [&]() -> float {
        bool hasU = (ia < NUM_STEPS), hasN = (ib < UP_STEPS);
        float zu = nearz + span * ((float)ia * (1.0f / 127.0f));
        float zn = hasN ? s_newz[ib * BLOCK + tid] : 3.0e38f;
        if (hasU && (!hasN || zu <= zn)) { ++ia; return zu; }
        ++ib; return zn;
    };

    T = 1.0f;
    float img0 = 0.f, img1 = 0.f, img2 = 0.f, wsum = 0.f;
    float zcur = pop();
#pragma unroll 1
    for (int s = 0; s < NUM_STEPS + UP_STEPS; ++s) {
        float znext = (s < NUM_STEPS + UP_STEPS - 1) ? pop() : 0.0f;
        float delta = (s < NUM_STEPS + UP_STEPS - 1) ? (znext - zcur) : sdist;

        float x0 = fminf(fmaxf(ox + dx * zcur, -1.0f), 1.0f);
        float x1 = fminf(fmaxf(oy + dy * zcur, -1.0f), 1.0f);
        float x2 = fminf(fmaxf(oz + dz * zcur, -1.0f), 1.0f);

        float e0 = x0 - cx, e1 = x1 - cy, e2 = x2 - cz;
        float d2 = e0 * e0 + e1 * e1 + e2 * e2;
        float sig = sA * __expf(-d2 * inv2s2);

        float alpha = 1.0f - __expf(-delta * sig);
        float w = alpha * T;
        T *= (1.0f - alpha + 1e-15f);

        float g0 = fmaf(x0, W[0], fmaf(x1, W[3], fmaf(x2, W[6], b0)));
        float g1 = fmaf(x0, W[1], fmaf(x1, W[4], fmaf(x2, W[7], b1)));
        float g2 = fmaf(x0, W[2], fmaf(x1, W[5], fmaf(x2, W[8], b2)));
        img0 += w * sigmoidf_(g0);
        img1 += w * sigmoidf_(g1);
        img2 += w * sigmoidf_(g2);
        wsum += w;
        zcur = znext;
    }

    const float bg = 1.0f - wsum;  // white background
    float* o = out + (size_t)(ray0 + tid) * 3;
    o[0] = img0 + bg;
    o[1] = img1 + bg;
    o[2] = img2 + bg;
}

extern "C" void kernel_launch(void* const* d_in, const int* in_sizes, int n_in,
                              void* d_out, int out_size, void* d_ws, size_t ws_size,
                              hipStream_t stream) {
    (void)n_in; (void)out_size; (void)d_ws; (void)ws_size;
    const float* rays_o = (const float*)d_in[0];
    const float* rays_d = (const float*)d_in[1];
    const float* c      = (const float*)d_in[2];
    const float* log_s  = (const float*)d_in[3];
    const float* amp    = (const float*)d_in[4];
    const float* Wc     = (const float*)d_in[5];
    const float* bc     = (const float*)d_in[6];
    const int n_rays = in_sizes[0] / 3;

    dim3 grid(n_rays / BLOCK), block(BLOCK);
    nerf_render_kernel<<<grid, block, 0, stream>>>(rays_o, rays_d, c, log_s, amp,
                                                   Wc, bc, (float*)d_out);
}